// Variable_Attention_layer_22187801051686
// MI455X (gfx1250) — compile-verified
//
#include <hip/hip_runtime.h>
#include <hip/hip_bf16.h>

#define BATCH 16
#define NSEQ  2048
#define TDIM  96
#define DDIM  128
#define SC_STRIDE 2052   // 2048 + 4 pad: shifts 8-row offset by 32 banks

typedef float v2f __attribute__((ext_vector_type(2)));
typedef float v4f __attribute__((ext_vector_type(4)));
typedef float v8f __attribute__((ext_vector_type(8)));

__device__ __forceinline__ v8f wmma_f32_k4(v2f a, v2f b, v8f c) {
    // D = A(16x4) * B(4x16) + C, fp32 WMMA
    return __builtin_amdgcn_wmma_f32_16x16x4_f32(
        /*neg_a=*/false, a, /*neg_b=*/false, b,
        /*c_mod=*/(short)0, c, /*reuse_a=*/false, /*reuse_b=*/false);
}

// ---------------------------------------------------------------------------
// Kernel A: Q = x @ WQ, K = x @ WK   (rows = B*N flattened, fp32 WMMA)
// grid = (B*N)/16 blocks, 256 threads (8 waves). Wave w does jobs w and w+8:
// job j: matrix = j/8 (Q or K), column tile = (j&7)*16.
// ---------------------------------------------------------------------------
__global__ __launch_bounds__(256) void qk_proj_kernel(
    const float* __restrict__ x, const float* __restrict__ WQ,
    const float* __restrict__ WK, float* __restrict__ Qo, float* __restrict__ Ko)
{
    __shared__ float xs[16 * TDIM];          // 6 KB x tile
    const int tid  = threadIdx.x;
    const int lane = tid & 31;
    const int wave = tid >> 5;
    const int ln16 = lane & 15;
    const int hi   = lane >> 4;              // 0 or 1 (half-wave)
    const long long r0 = (long long)blockIdx.x * 16;

    // x-tile rows are contiguous in memory: copy 16*96 floats straight in
    for (int i = tid; i < 16 * TDIM; i += 256)
        xs[i] = x[r0 * TDIM + i];
    __syncthreads();

    // A-operand base: one LDS pointer, immediate offsets (ds_load_b64)
    const v2f* xp = (const v2f*)(xs + ln16 * TDIM + hi * 2); // v2f idx: kk*2

    for (int j = wave; j < 16; j += 8) {     // wave-uniform: EXEC stays full
        const float* W   = (j < 8) ? WQ : WK;
        float*       Out = (j < 8) ? Qo : Ko;
        const int col0 = (j & 7) * 16;
        // B-operand base: W[2hi][col0+ln16]; b.x at +kk*512, b.y at +128 more
        const float* wp = W + (size_t)hi * 2 * DDIM + col0 + ln16;
        v8f acc = {};
        #pragma unroll
        for (int kk = 0; kk < TDIM / 4; ++kk) {
            v2f a = xp[kk * 2];              // x[row][4kk+2hi .. +1]
            v2f b;
            b.x = wp[kk * 512];              // W[4kk+2hi  ][col]
            b.y = wp[kk * 512 + DDIM];       // W[4kk+2hi+1][col]
            acc = wmma_f32_k4(a, b, acc);
        }
        #pragma unroll
        for (int v = 0; v < 8; ++v) {
            const int m = v + 8 * hi;        // C layout: vgpr v, half-wave hi
            Out[(r0 + m) * DDIM + col0 + ln16] = acc[v];
        }
    }
}

// ---------------------------------------------------------------------------
// Kernel B: per (batch, 16-row tile): scores = relu(Q K^T)/sqrt(D), softmax.
// grid = B * (N/16) = 2048 blocks, 256 threads, ~132 KB dynamic LDS.
// ---------------------------------------------------------------------------
__global__ __launch_bounds__(256) void attn_softmax_kernel(
    const float* __restrict__ Q, const float* __restrict__ K,
    float* __restrict__ out)
{
    extern __shared__ float smem[];
    float* sc     = smem;                          // 16 x SC_STRIDE scores
    float* red    = smem + 16 * SC_STRIDE;         // 16 x 16 reduction
    float* rowmax = red + 256;
    float* rowinv = rowmax + 16;

    const int tid  = threadIdx.x;
    const int lane = tid & 31;
    const int wave = tid >> 5;
    const int ln16 = lane & 15;
    const int hi   = lane >> 4;

    const int b  = blockIdx.x >> 7;                // 128 row-tiles per batch
    const int rt = blockIdx.x & 127;
    const long long rowbase = (long long)b * NSEQ + rt * 16;

    // Q tile -> registers in WMMA A layout (reused for all 128 column tiles)
    // single base pointer, v2f (b64) loads at immediate offsets
    const v2f* qp = (const v2f*)(Q + (rowbase + ln16) * DDIM + hi * 2);
    v2f qreg[32];
    #pragma unroll
    for (int kk = 0; kk < 32; ++kk)
        qreg[kk] = qp[kk * 2];

    const float scale = 0.08838834764831845f;      // 1/sqrt(128)
    const float* kb = K + (long long)b * NSEQ * DDIM;

    for (int ct = wave; ct < NSEQ / 16; ct += 8) { // wave-uniform loop
        const int col0 = ct * 16;
        // B-operand: K[col0+ln16][4kk+2hi .. +1] = contiguous v2f, b64 loads
        const v2f* krow = (const v2f*)(kb + (size_t)(col0 + ln16) * DDIM + hi * 2);
        v8f acc = {};
        #pragma unroll
        for (int kk = 0; kk < 32; ++kk)            // D=128 = 32 x K4
            acc = wmma_f32_k4(qreg[kk], krow[kk * 2], acc);
        #pragma unroll
        for (int v = 0; v < 8; ++v) {
            const int m = v + 8 * hi;
            float s = acc[v];
            s = (s > 0.0f ? s : 0.0f) * scale;     // relu + 1/sqrt(d)
            sc[m * SC_STRIDE + col0 + ln16] = s;
        }
    }
    __syncthreads();

    // ---- softmax over each 2048-wide row (16 threads per row) ----
    const int row  = tid >> 4;
    const int part = tid & 15;
    {
        float m = -1e30f;
        const float* p = &sc[row * SC_STRIDE];
        for (int c = part; c < NSEQ; c += 16) m = fmaxf(m, p[c]);
        red[row * 16 + part] = m;
    }
    __syncthreads();
    if (tid < 16) {
        float m = red[tid * 16];
        #pragma unroll
        for (int i = 1; i < 16; ++i) m = fmaxf(m, red[tid * 16 + i]);
        rowmax[tid] = m;
    }
    __syncthreads();
    {
        const float rm = rowmax[row];
        float s = 0.0f;
        float* p = &sc[row * SC_STRIDE];
        for (int c = part; c < NSEQ; c += 16) {
            const float e = __expf(p[c] - rm);
            p[c] = e;
            s += e;
        }
        red[row * 16 + part] = s;
    }
    __syncthreads();
    if (tid < 16) {
        float s = 0.0f;
        #pragma unroll
        for (int i = 0; i < 16; ++i) s += red[tid * 16 + i];
        rowinv[tid] = 1.0f / s;
    }
    __syncthreads();

    // ---- normalized, float4-coalesced, NON-TEMPORAL 256 MB store ----
    // (write-once stream: keep Q/K resident in L2 instead of rinsing it)
    for (int i = tid; i < 16 * (NSEQ / 4); i += 256) {
        const int r  = i >> 9;                     // NSEQ/4 = 512
        const int c4 = (i & 511) * 4;
        const v4f v = *reinterpret_cast<const v4f*>(&sc[r * SC_STRIDE + c4]);
        const float inv = rowinv[r];
        v4f o = v * inv;
        __builtin_nontemporal_store(o,
            reinterpret_cast<v4f*>(&out[(rowbase + r) * (long long)NSEQ + c4]));
    }
}

// ---------------------------------------------------------------------------
extern "C" void kernel_launch(void* const* d_in, const int* in_sizes, int n_in,
                              void* d_out, int out_size, void* d_ws, size_t ws_size,
                              hipStream_t stream) {
    const float* x  = (const float*)d_in[0];   // (B, N, T)
    const float* WQ = (const float*)d_in[1];   // (T, D)
    const float* WK = (const float*)d_in[2];   // (T, D)
    float* out = (float*)d_out;                // (B, N, N)

    float* Qw = (float*)d_ws;                                  // 16 MB
    float* Kw = Qw + (size_t)BATCH * NSEQ * DDIM;              // 16 MB

    const int tiles = (BATCH * NSEQ) / 16;     // 2048
    qk_proj_kernel<<<tiles, 256, 0, stream>>>(x, WQ, WK, Qw, Kw);

    const size_t shmem = (size_t)(16 * SC_STRIDE + 256 + 16 + 16) * sizeof(float);
    attn_softmax_kernel<<<BATCH * (NSEQ / 16), 256, shmem, stream>>>(Qw, Kw, out);
}